// GCNEncoder_37726992728721
// MI455X (gfx1250) — compile-verified
//
#include <hip/hip_runtime.h>
#include <hip/hip_bf16.h>

typedef __attribute__((ext_vector_type(2))) float v2f;
typedef __attribute__((ext_vector_type(8))) float v8f;

#define DFEAT 64

// ---------------- elementwise / degree kernels ----------------

__global__ void zero_f32_kernel(float* __restrict__ p, int n) {
    int i = blockIdx.x * blockDim.x + threadIdx.x;
    if (i < n) p[i] = 0.0f;
}

__global__ void degree_kernel(const int* __restrict__ dst, float* __restrict__ deg, int E) {
    int e = blockIdx.x * blockDim.x + threadIdx.x;
    if (e < E) atomicAdd(&deg[dst[e]], 1.0f);
}

__global__ void dinv_kernel(const float* __restrict__ deg, float* __restrict__ dinv, int n) {
    int i = blockIdx.x * blockDim.x + threadIdx.x;
    if (i < n) dinv[i] = rsqrtf(deg[i] + 1.0f);
}

// agg[i,j] = h[i,j] * dinv[i]^2   (self-loop term init)
__global__ void selfloop_init_kernel(const float* __restrict__ h, const float* __restrict__ dinv,
                                     float* __restrict__ agg, int total) {
    int idx = blockIdx.x * blockDim.x + threadIdx.x;
    if (idx >= total) return;
    int node = idx >> 6;
    float di = dinv[node];
    agg[idx] = h[idx] * (di * di);
}

// out[i,j] = h[i,j] * dinv[i]^2 + b[j]   (self-loop + bias init for layer 2)
__global__ void selfloop_bias_init_kernel(const float* __restrict__ h, const float* __restrict__ dinv,
                                          const float* __restrict__ b, float* __restrict__ out, int total) {
    int idx = blockIdx.x * blockDim.x + threadIdx.x;
    if (idx >= total) return;
    int node = idx >> 6;
    int j = idx & (DFEAT - 1);
    float di = dinv[node];
    out[idx] = h[idx] * (di * di) + b[j];
}

// x[i,j] = relu(agg[i,j] + b[j])   (in place)
__global__ void relu_bias_kernel(float* __restrict__ agg, const float* __restrict__ b, int total) {
    int idx = blockIdx.x * blockDim.x + threadIdx.x;
    if (idx >= total) return;
    int j = idx & (DFEAT - 1);
    float v = agg[idx] + b[j];
    agg[idx] = v > 0.0f ? v : 0.0f;
}

// Edge scatter: 16 threads per edge, 4 floats each (float4 gather, 4x atomic f32 scatter)
__global__ void edge_scatter_kernel(const int* __restrict__ src, const int* __restrict__ dst,
                                    const float* __restrict__ dinv, const float* __restrict__ h,
                                    float* __restrict__ agg, int E) {
    int tid = blockIdx.x * blockDim.x + threadIdx.x;
    int e = tid >> 4;
    if (e >= E) return;
    int j = (tid & 15) << 2;
    int s = src[e];
    int d = dst[e];
    float c = dinv[s] * dinv[d];
    const float4 hv = *(const float4*)(h + (size_t)s * DFEAT + j);
    float* ap = agg + (size_t)d * DFEAT + j;
    atomicAdd(ap + 0, hv.x * c);
    atomicAdd(ap + 1, hv.y * c);
    atomicAdd(ap + 2, hv.z * c);
    atomicAdd(ap + 3, hv.w * c);
}

// ---------------- WMMA f32 GEMM: out[n,64] = X[n,64] @ W[64,64] ----------------
// One block = 128 threads = 4 waves; each wave computes a 16x64 output tile via
// 4 accumulators of V_WMMA_F32_16X16X4_F32, K swept in 16 steps of 4.
// A (16x4 f32): lanes 0-15 hold K=k0,k0+1; lanes 16-31 hold K=k0+2,k0+3 (VGPR0/1).
// B (4x16 f32): same K halves, one output column per lane group.
// C/D (16x16 f32, v8f): VGPR v -> row M=v (lanes 0-15) / M=v+8 (lanes 16-31), N=lane&15.
__global__ __launch_bounds__(128) void gemm64_wmma_kernel(const float* __restrict__ X,
                                                          const float* __restrict__ W,
                                                          float* __restrict__ out, int n) {
    __shared__ float Wl[DFEAT * DFEAT];

    const int t = threadIdx.x;
    // Cooperative load of the 16 KB weight matrix into LDS (float4 = 8 per thread).
    {
        const float4* Wv = (const float4*)W;
        float4* Wlv = (float4*)Wl;
        #pragma unroll
        for (int i = 0; i < (DFEAT * DFEAT / 4) / 128; ++i)
            Wlv[t + i * 128] = Wv[t + i * 128];
    }
    __syncthreads();

    const int wave = t >> 5;
    const int lane = t & 31;
    const int half = lane >> 4;   // 0: lanes 0-15, 1: lanes 16-31
    const int l16 = lane & 15;
    const int rowBase = blockIdx.x * 64 + wave * 16;

    v8f acc[4] = {};

    // A-fragment row for this lane (clamped; wave-uniform control flow for WMMA)
    int rA = rowBase + l16;
    if (rA > n - 1) rA = n - 1;
    const float* Xrow = X + (size_t)rA * DFEAT;

    for (int k0 = 0; k0 < DFEAT; k0 += 4) {
        const int ka = k0 + half * 2;
        v2f a;
        a[0] = Xrow[ka];
        a[1] = Xrow[ka + 1];
        #pragma unroll
        for (int tn = 0; tn < 4; ++tn) {
            const int n0 = tn * 16;
            v2f b;
            b[0] = Wl[ka * DFEAT + n0 + l16];
            b[1] = Wl[(ka + 1) * DFEAT + n0 + l16];
            acc[tn] = __builtin_amdgcn_wmma_f32_16x16x4_f32(
                /*neg_a=*/false, a, /*neg_b=*/false, b,
                /*c_mod=*/(short)0, acc[tn], /*reuse_a=*/false, /*reuse_b=*/false);
        }
    }

    // Store D tiles
    #pragma unroll
    for (int tn = 0; tn < 4; ++tn) {
        const int n0 = tn * 16;
        #pragma unroll
        for (int v = 0; v < 8; ++v) {
            int row = rowBase + v + half * 8;
            if (row < n) out[(size_t)row * DFEAT + n0 + l16] = acc[tn][v];
        }
    }
}

// ---------------- launch ----------------

extern "C" void kernel_launch(void* const* d_in, const int* in_sizes, int n_in,
                              void* d_out, int out_size, void* d_ws, size_t ws_size,
                              hipStream_t stream) {
    const int* eidx = (const int*)d_in[0];          // prop_edge_index, shape (2, E)
    const float* emb = (const float*)d_in[1];       // (N, 64)
    const float* W1 = (const float*)d_in[2];        // (64, 64)
    const float* b1 = (const float*)d_in[3];        // (64,)
    const float* W2 = (const float*)d_in[4];        // (64, 64)
    const float* b2 = (const float*)d_in[5];        // (64,)
    float* out = (float*)d_out;                     // (N, 64)

    const int E = in_sizes[0] / 2;
    const int n = in_sizes[1] / DFEAT;
    const int total = n * DFEAT;

    const int* src = eidx;
    const int* dst = eidx + E;

    // workspace layout (floats)
    float* deg  = (float*)d_ws;
    float* dinv = deg + n;
    float* h1   = dinv + n;
    float* agg1 = h1 + (size_t)n * DFEAT;           // becomes x after relu
    float* h2   = agg1 + (size_t)n * DFEAT;

    const int B = 256;
    const int gN = (n + B - 1) / B;
    const int gE = (E + B - 1) / B;
    const int gT = (total + B - 1) / B;
    const int gScatter = (E * 16 + B - 1) / B;
    const int gGemm = (n + 63) / 64;

    // degree + normalization
    zero_f32_kernel<<<gN, B, 0, stream>>>(deg, n);
    degree_kernel<<<gE, B, 0, stream>>>(dst, deg, E);
    dinv_kernel<<<gN, B, 0, stream>>>(deg, dinv, n);

    // layer 1
    gemm64_wmma_kernel<<<gGemm, 128, 0, stream>>>(emb, W1, h1, n);
    selfloop_init_kernel<<<gT, B, 0, stream>>>(h1, dinv, agg1, total);
    edge_scatter_kernel<<<gScatter, B, 0, stream>>>(src, dst, dinv, h1, agg1, E);
    relu_bias_kernel<<<gT, B, 0, stream>>>(agg1, b1, total);

    // layer 2
    gemm64_wmma_kernel<<<gGemm, 128, 0, stream>>>(agg1, W2, h2, n);
    selfloop_bias_init_kernel<<<gT, B, 0, stream>>>(h2, dinv, b2, out, total);
    edge_scatter_kernel<<<gScatter, B, 0, stream>>>(src, dst, dinv, h2, out, E);
}